// JNetBlur_19284403159137
// MI455X (gfx1250) — compile-verified
//
#include <hip/hip_runtime.h>
#include <math.h>
#include <stdint.h>

// ---------------------------------------------------------------------------
// JNetBlur for MI455X (gfx1250): separable 9x9x9 Gaussian blur, stride-4 in z,
// global min-max normalize.
//   Pass 1: strided z-conv (float4 vectorized, VALU)           inp -> ws
//   Pass 2: fused x/y convs as banded matmuls on the f32 WMMA pipe
//           (V_WMMA_F32_16X16X4_F32), halo staged to LDS with
//           global_load_async_to_lds_b128 on interior blocks, with a fused
//           per-block min/max partial reduction                ws -> out
//   Pass 3: final min/max reduce over 4096 partials
//   Pass 4: in-place normalize (float4)
// ---------------------------------------------------------------------------

typedef __attribute__((ext_vector_type(2))) float v2f;
typedef __attribute__((ext_vector_type(8))) float v8f;

#define D_IN   128
#define HH     512
#define WW     512
#define D_OUT  32
#define PLANE  (HH * WW)          // 262144
#define N_OUT  (D_OUT * PLANE)    // 8388608
#define INV_SQRT_2PI 0.3989422804014327f

#define XS_STRIDE 140             // halo row stride: 560 B = 35*16 (B128-aligned)
#define MS_STRIDE 17              // per-wave mid row stride (24 x 16 payload)
#define NBLK2 (4 * 32 * 32)       // xyconv grid size = #minmax partials (4096)

// ---- Pass 1: strided z-conv (inp -> ws).  Weights folded with z0*inv_sqrt_2pi.
__global__ void zconv_kernel(const float4* __restrict__ inp,
                             const float* __restrict__ mu_z,
                             const float* __restrict__ sig_z,
                             const float* __restrict__ bet_z,
                             float4* __restrict__ out1) {
    __shared__ float fz[9];
    if (threadIdx.x < 9) {
        float z0 = expf(mu_z[0] + 0.5f * sig_z[0] * sig_z[0]);
        float t  = (float)((int)threadIdx.x - 4);
        float b  = bet_z[0];
        fz[threadIdx.x] = expf(-(t * t) / (2.f * b * b)) * z0 * INV_SQRT_2PI;
    }
    __syncthreads();
    int idx = blockIdx.x * 256 + threadIdx.x;      // over N_OUT/4 float4s
    if (idx >= N_OUT / 4) return;
    int oz  = idx >> 16;                           // / (PLANE/4)
    int rem = idx & (PLANE / 4 - 1);
    int zb  = 4 * oz - 3;                          // pad_z = 3, stride 4
    float4 s = {0.f, 0.f, 0.f, 0.f};
#pragma unroll
    for (int k = 0; k < 9; ++k) {
        int z = zb + k;
        if (0 <= z && z < D_IN) {
            float4 v = inp[z * (PLANE / 4) + rem];
            float f = fz[k];
            s.x = fmaf(f, v.x, s.x);
            s.y = fmaf(f, v.y, s.y);
            s.z = fmaf(f, v.z, s.z);
            s.w = fmaf(f, v.w, s.w);
        }
    }
    out1[idx] = s;
}

// ---- Pass 2: fused x/y 9-tap convs.  256 threads = 8 waves; block covers a
// 16(h) x 128(w) output strip of one z-slice; each wave owns a 16x16 tile.
//   step A:  mid(24x16) = X(24x24) * S(24x16),  S[c][n] = fxy[c-n]   (W-conv)
//   step B:  out(16x16) = T(16x24) * mid(24x16), T[m][r] = fxy[r-m]  (H-conv)
__global__ void xyconv_wmma_kernel(const float* __restrict__ zin,
                                   const float* __restrict__ bet_xy,
                                   float* __restrict__ out,
                                   float* __restrict__ pmin,
                                   float* __restrict__ pmax) {
    __shared__ float fw[9];
    __shared__ __align__(16) float Xs[24 * XS_STRIDE];  // halo tile 24 x 136
    __shared__ float Ms[8 * 24 * MS_STRIDE];            // per-wave mid buffers

    const int tid  = threadIdx.x;
    const int wv   = tid >> 5;                // wave 0..7
    const int lane = tid & 31;
    const int kh   = lane >> 4;               // lane half (K pair / M half)
    const int mn   = lane & 15;

    if (tid < 9) {
        float t = (float)(tid - 4);
        float b = bet_xy[0];
        fw[tid] = expf(-(t * t) / (2.f * b * b));
    }

    const int w0 = blockIdx.x * 128;          // strip origin (w)
    const int h0 = blockIdx.y * 16;           // strip origin (h)
    const int z  = blockIdx.z;
    const float* plane = zin + (size_t)z * PLANE;

    // ---- stage the 24 x 136 halo tile into LDS
    const bool interior = (h0 >= 4) && (h0 + 20 <= HH) &&
                          (w0 >= 4) && (w0 + 132 <= WW);
    if (interior) {
        // gfx1250 async copy, 16 B per lane: global addr -> per-lane LDS offset.
        // Row starts are 16B aligned in memory ((w0-4)*4 is a multiple of 16)
        // and in LDS (XS_STRIDE*4 = 560 = 35*16).
        for (int i = tid; i < 24 * 34; i += 256) {        // 34 float4 per row
            int r = i / 34, c4 = i % 34;
            const float* gsrc = plane + (h0 - 4 + r) * WW + (w0 - 4) + 4 * c4;
            uint64_t gaddr  = (uint64_t)(uintptr_t)gsrc;
            uint32_t ldsoff = (uint32_t)(uintptr_t)&Xs[r * XS_STRIDE + 4 * c4];
            asm volatile("global_load_async_to_lds_b128 %0, %1, off"
                         :: "v"(ldsoff), "v"(gaddr) : "memory");
        }
    } else {
        for (int i = tid; i < 24 * 136; i += 256) {
            int r = i / 136, c = i % 136;
            int gh = h0 - 4 + r, gw = w0 - 4 + c;
            float v = 0.f;
            if (gh >= 0 && gh < HH && gw >= 0 && gw < WW) v = plane[gh * WW + gw];
            Xs[r * XS_STRIDE + c] = v;
        }
    }
    asm volatile("s_wait_asynccnt 0x0" ::: "memory");
    __syncthreads();

    auto band = [&](int d) -> float { return ((unsigned)d <= 8u) ? fw[d] : 0.f; };

    const int cw = wv * 16;                   // wave's column base inside halo
    float* Mw = Ms + wv * 24 * MS_STRIDE;

    // ---- step A: two M-tiles over the 24 halo rows (rows 0-15 and 16-23)
    v8f midlo = {};
    v8f midhi = {};
#pragma unroll
    for (int k0 = 0; k0 < 24; k0 += 4) {
        const int kr = k0 + 2 * kh;
        v2f b;  b.x = band(kr - mn);  b.y = band(kr + 1 - mn);   // S rows kr,kr+1
        v2f al;                                                  // X row mn
        al.x = Xs[mn * XS_STRIDE + cw + kr];
        al.y = Xs[mn * XS_STRIDE + cw + kr + 1];
        midlo = __builtin_amdgcn_wmma_f32_16x16x4_f32(false, al, false, b,
                                                      (short)0, midlo, false, false);
        const int rhi = 16 + mn;                                 // X row 16+mn
        v2f ah;
        ah.x = (rhi < 24) ? Xs[rhi * XS_STRIDE + cw + kr]     : 0.f;
        ah.y = (rhi < 24) ? Xs[rhi * XS_STRIDE + cw + kr + 1] : 0.f;
        midhi = __builtin_amdgcn_wmma_f32_16x16x4_f32(false, ah, false, b,
                                                      (short)0, midhi, false, false);
    }

    // spill mid (24x16) to LDS to re-stripe from C/D layout into B layout
#pragma unroll
    for (int v = 0; v < 8; ++v) {
        int r = v + 8 * kh;                   // C/D: VGPR v holds rows v / v+8
        Mw[r * MS_STRIDE + mn] = midlo[v];
        int r2 = 16 + r;
        if (r2 < 24) Mw[r2 * MS_STRIDE + mn] = midhi[v];
    }
    __syncthreads();

    // ---- step B: out = T(16x24) * mid(24x16)
    v8f acc = {};
#pragma unroll
    for (int k0 = 0; k0 < 24; k0 += 4) {
        const int kr = k0 + 2 * kh;
        v2f a; a.x = band(kr - mn);  a.y = band(kr + 1 - mn);    // T row mn
        v2f b; b.x = Mw[kr * MS_STRIDE + mn];                    // mid rows
               b.y = Mw[(kr + 1) * MS_STRIDE + mn];
        acc = __builtin_amdgcn_wmma_f32_16x16x4_f32(false, a, false, b,
                                                    (short)0, acc, false, false);
    }

    float* oplane = out + (size_t)z * PLANE;
    float mn_l = 3.402823466e38f, mx_l = -3.402823466e38f;
#pragma unroll
    for (int v = 0; v < 8; ++v) {
        int m = v + 8 * kh;
        float val = acc[v];
        oplane[(h0 + m) * WW + (w0 + cw + mn)] = val;
        mn_l = fminf(mn_l, val);
        mx_l = fmaxf(mx_l, val);
    }

    // ---- fused per-block min/max partial (reuse Xs as scratch)
    __syncthreads();                          // all tile reads of Xs/Ms done
    float* smn = Xs;
    float* smx = Xs + 256;
    smn[tid] = mn_l;
    smx[tid] = mx_l;
    __syncthreads();
#pragma unroll
    for (int s = 128; s > 0; s >>= 1) {
        if (tid < s) {
            smn[tid] = fminf(smn[tid], smn[tid + s]);
            smx[tid] = fmaxf(smx[tid], smx[tid + s]);
        }
        __syncthreads();
    }
    if (tid == 0) {
        int bid = blockIdx.x + gridDim.x * (blockIdx.y + gridDim.y * blockIdx.z);
        pmin[bid] = smn[0];
        pmax[bid] = smx[0];
    }
}

// ---- Pass 3: single-block final reduce -> mnmx[0]=min, mnmx[1]=max
__global__ void minmax_final(const float* __restrict__ pmin,
                             const float* __restrict__ pmax,
                             int n, float* __restrict__ mnmx) {
    __shared__ float smn[256], smx[256];
    float mn = 3.402823466e38f, mx = -3.402823466e38f;
    for (int i = threadIdx.x; i < n; i += 256) {
        mn = fminf(mn, pmin[i]);
        mx = fmaxf(mx, pmax[i]);
    }
    smn[threadIdx.x] = mn; smx[threadIdx.x] = mx;
    __syncthreads();
#pragma unroll
    for (int s = 128; s > 0; s >>= 1) {
        if (threadIdx.x < s) {
            smn[threadIdx.x] = fminf(smn[threadIdx.x], smn[threadIdx.x + s]);
            smx[threadIdx.x] = fmaxf(smx[threadIdx.x], smx[threadIdx.x + s]);
        }
        __syncthreads();
    }
    if (threadIdx.x == 0) { mnmx[0] = smn[0]; mnmx[1] = smx[0]; }
}

// ---- Pass 4: in-place normalize (d_out fully rewritten by pass 2 each call)
__global__ void normalize_kernel(float4* __restrict__ out,
                                 const float* __restrict__ mnmx) {
    int idx = blockIdx.x * 256 + threadIdx.x;
    if (idx >= N_OUT / 4) return;
    float mn = mnmx[0];
    float r  = 1.f / (mnmx[1] - mn);
    float4 v = out[idx];
    v.x = (v.x - mn) * r;
    v.y = (v.y - mn) * r;
    v.z = (v.z - mn) * r;
    v.w = (v.w - mn) * r;
    out[idx] = v;
}

extern "C" void kernel_launch(void* const* d_in, const int* in_sizes, int n_in,
                              void* d_out, int out_size, void* d_ws, size_t ws_size,
                              hipStream_t stream) {
    const float* inp   = (const float*)d_in[0];
    const float* mu_z  = (const float*)d_in[1];
    const float* sig_z = (const float*)d_in[2];
    const float* bxy   = (const float*)d_in[3];
    const float* bz    = (const float*)d_in[4];
    float* out  = (float*)d_out;

    float* ws1  = (float*)d_ws;        // N_OUT floats: z-conv intermediate
    float* pmin = ws1 + N_OUT;         // NBLK2 partials
    float* pmax = pmin + NBLK2;        // NBLK2 partials
    float* mnmx = pmax + NBLK2;        // 2 scalars

    zconv_kernel<<<(N_OUT / 4 + 255) / 256, 256, 0, stream>>>(
        (const float4*)inp, mu_z, sig_z, bz, (float4*)ws1);

    dim3 g2(WW / 128, HH / 16, D_OUT);                 // 4 x 32 x 32 = 4096
    xyconv_wmma_kernel<<<g2, 256, 0, stream>>>(ws1, bxy, out, pmin, pmax);

    minmax_final<<<1, 256, 0, stream>>>(pmin, pmax, NBLK2, mnmx);

    normalize_kernel<<<(N_OUT / 4 + 255) / 256, 256, 0, stream>>>(
        (float4*)out, mnmx);
}